// MulticoreBPFLayer_17832704213311
// MI455X (gfx1250) — compile-verified
//
#include <hip/hip_runtime.h>
#include <hip/hip_bf16.h>

typedef __attribute__((ext_vector_type(2))) float v2f;
typedef __attribute__((ext_vector_type(8))) float v8f;

#define PN 8192
#define CN 8192
#define KCHUNK 1024

// workspace layout (float offsets)
#define OFF_STATE 0        // 8192*3 = 24576
#define OFF_VS    24576    // 8192*4 = 32768  (v0,v1,v2,s per row)
#define OFF_W     57344    // 8192
#define OFF_PREF  65536    // 8192
#define OFF_G     73728    // 9 (3x3 Gram)
#define OFF_ACC   73737    // 3
#define OFF_WTOT  73740    // 1

__device__ __forceinline__ unsigned pcg_hash(unsigned x) {
    x = x * 747796405u + 2891336453u;
    unsigned w = ((x >> ((x >> 28u) + 4u)) ^ x) * 277803737u;
    return (w >> 22u) ^ w;
}
__device__ __forceinline__ float u01(unsigned h) {
    return ((float)(h >> 8) + 0.5f) * (1.0f / 16777216.0f);
}

// ---- zero the accumulation buffers (harness poisons ws) ----
__global__ void k_zero(float* ws) {
    int i = blockIdx.x * blockDim.x + threadIdx.x;
    if (i < 32768) ws[OFF_VS + i] = 0.0f;
    if (i < 16)    ws[OFF_G + i]  = 0.0f;   // covers G(9)+acc(3)+wtot(1)+pad
}

// ---- state transition: x' = T x + L z, L = chol(Q), z ~ N(0,1) (hash RNG) ----
__global__ void k_state(const float* __restrict__ sv, const float* __restrict__ T,
                        const float* __restrict__ Q, float* __restrict__ ws) {
    int p = blockIdx.x * blockDim.x + threadIdx.x;
    if (p >= PN) return;
    // 3x3 Cholesky in registers (redundant per thread; trivially cheap)
    float l00 = sqrtf(Q[0]);
    float l10 = Q[3] / l00, l20 = Q[6] / l00;
    float l11 = sqrtf(Q[4] - l10 * l10);
    float l21 = (Q[7] - l20 * l10) / l11;
    float l22 = sqrtf(Q[8] - l20 * l20 - l21 * l21);

    float x0 = sv[p * 3 + 0], x1 = sv[p * 3 + 1], x2 = sv[p * 3 + 2];
    // (x @ T^T)[c] = sum_k x_k * T[c][k]
    float u0 = x0 * T[0] + x1 * T[1] + x2 * T[2];
    float u1 = x0 * T[3] + x1 * T[4] + x2 * T[5];
    float u2 = x0 * T[6] + x1 * T[7] + x2 * T[8];

    unsigned h0 = pcg_hash(0x1234567u ^ (p * 4u + 0u));
    unsigned h1 = pcg_hash(0x1234567u ^ (p * 4u + 1u));
    unsigned h2 = pcg_hash(0x1234567u ^ (p * 4u + 2u));
    unsigned h3 = pcg_hash(0x1234567u ^ (p * 4u + 3u));
    float ua = u01(h0), ub = u01(h1), uc = u01(h2), ud = u01(h3);
    float r0 = sqrtf(-2.0f * logf(ua));
    float r1 = sqrtf(-2.0f * logf(uc));
    float z0 = r0 * cosf(6.2831853f * ub);
    float z1 = r0 * sinf(6.2831853f * ub);
    float z2 = r1 * cosf(6.2831853f * ud);
    // (z @ L)[c] with L lower-triangular
    float e0 = z0 * l00 + z1 * l10 + z2 * l20;
    float e1 = z1 * l11 + z2 * l21;
    float e2 = z2 * l22;

    ws[OFF_STATE + p * 3 + 0] = u0 + e0;
    ws[OFF_STATE + p * 3 + 1] = u1 + e1;
    ws[OFF_STATE + p * 3 + 2] = u2 + e2;
}

// ---- G = H^T H (3x3, symmetric) ----
__global__ void k_gram(const float* __restrict__ fwd, float* __restrict__ ws) {
    float g00 = 0, g01 = 0, g02 = 0, g11 = 0, g12 = 0, g22 = 0;
    for (int j = blockIdx.x * blockDim.x + threadIdx.x; j < CN; j += gridDim.x * blockDim.x) {
        float f0 = fwd[j * 3 + 0], f1 = fwd[j * 3 + 1], f2 = fwd[j * 3 + 2];
        g00 += f0 * f0; g01 += f0 * f1; g02 += f0 * f2;
        g11 += f1 * f1; g12 += f1 * f2; g22 += f2 * f2;
    }
    atomicAdd(&ws[OFF_G + 0], g00);
    atomicAdd(&ws[OFF_G + 1], g01);
    atomicAdd(&ws[OFF_G + 2], g02);
    atomicAdd(&ws[OFF_G + 4], g11);
    atomicAdd(&ws[OFF_G + 5], g12);
    atomicAdd(&ws[OFF_G + 8], g22);
}

// ---- main streaming pass: per-row  v = inputs @ H  (cols 0..2)  and  s = row sum of squares (col 3)
// One wave owns a 16-row tile; K split over blockIdx.y in chunks of KCHUNK.
// WMMA_F32_16X16X4_F32, two independent accumulator chains (v in accV, s in accS).
__global__ void __launch_bounds__(128)
k_main(const float* __restrict__ inp, const float* __restrict__ fwd, float* __restrict__ ws) {
    __shared__ float bs[KCHUNK * 4 + 16];     // H rows padded to 4 floats: [f0,f1,f2,0]
    const int kbase = blockIdx.y * KCHUNK;
    for (int jj = threadIdx.x; jj < KCHUNK; jj += 128) {
        int j = kbase + jj;
        bs[jj * 4 + 0] = fwd[j * 3 + 0];
        bs[jj * 4 + 1] = fwd[j * 3 + 1];
        bs[jj * 4 + 2] = fwd[j * 3 + 2];
        bs[jj * 4 + 3] = 0.0f;
    }
    if (threadIdx.x < 16) bs[KCHUNK * 4 + threadIdx.x] = 0.0f;
    __syncthreads();

    const int wave = threadIdx.x >> 5;
    const int lane = threadIdx.x & 31;
    const int m    = lane & 15;        // A: row within tile; B/C: column n
    const int half = lane >> 4;        // selects K pair (A/B) / row half (C)
    const int n    = m;
    const int rowtile = blockIdx.x * 4 + wave;
    const size_t row  = (size_t)rowtile * 16 + m;

    const float* ap = inp + row * (size_t)CN + kbase + 2 * half;
    const bool  bvalid = (n < 4);
    const float ones3  = (n == 3) ? 1.0f : 0.0f;   // B2: only column 3 is ones
    v2f b2v; b2v.x = ones3; b2v.y = ones3;

    v8f accV = {0.f, 0.f, 0.f, 0.f, 0.f, 0.f, 0.f, 0.f};  // cols 0..2 (col 3 stays 0)
    v8f accS = {0.f, 0.f, 0.f, 0.f, 0.f, 0.f, 0.f, 0.f};  // col 3 only (cols 0..2 stay 0)

    for (int kk = 0; kk < KCHUNK; kk += 4) {
        v2f a = *(const v2f*)(ap + kk);                 // A layout: VGPR0=K(2h), VGPR1=K(2h+1)
        __builtin_prefetch(ap + kk + 128, 0, 0);        // global_prefetch_b8
        int bi = (kk + 2 * half) * 4 + n;               // B layout mirrors A across lane halves
        float bb0 = bs[bi];
        float bb1 = bs[bi + 4];
        v2f b1;
        b1.x = bvalid ? bb0 : 0.0f;
        b1.y = bvalid ? bb1 : 0.0f;
        v2f asq; asq.x = a.x * a.x; asq.y = a.y * a.y;
        accV = __builtin_amdgcn_wmma_f32_16x16x4_f32(false, a,   false, b1,  (short)0, accV, false, false);
        accS = __builtin_amdgcn_wmma_f32_16x16x4_f32(false, asq, false, b2v, (short)0, accS, false, false);
    }

    // C layout: lane(n, half), VGPR r -> row M = 8*half + r, col n.
    // accV col3 == 0 and accS cols 0..2 == 0, so their sum is the combined tile.
    if (n < 4) {
        float* vs = ws + OFF_VS;
        #pragma unroll
        for (int r = 0; r < 8; ++r) {
            int rowm = rowtile * 16 + 8 * half + r;
            atomicAdd(&vs[rowm * 4 + n], accV[r] + accS[r]);
        }
    }
}

// ---- w[i] = s - 2 v.x + x^T G x ----
__global__ void k_weights(float* __restrict__ ws) {
    int i = blockIdx.x * blockDim.x + threadIdx.x;
    if (i >= PN) return;
    const float* st = ws + OFF_STATE;
    const float* vs = ws + OFF_VS;
    const float* G  = ws + OFF_G;
    float x0 = st[i * 3 + 0], x1 = st[i * 3 + 1], x2 = st[i * 3 + 2];
    float v0 = vs[i * 4 + 0], v1 = vs[i * 4 + 1], v2 = vs[i * 4 + 2], s = vs[i * 4 + 3];
    float g00 = G[0], g01 = G[1], g02 = G[2], g11 = G[4], g12 = G[5], g22 = G[8];
    float q = g00 * x0 * x0 + g11 * x1 * x1 + g22 * x2 * x2
            + 2.0f * (g01 * x0 * x1 + g02 * x0 * x2 + g12 * x1 * x2);
    float w = s - 2.0f * (v0 * x0 + v1 * x1 + v2 * x2) + q;
    ws[OFF_W + i] = fmaxf(w, 0.0f);   // probabilities proportional to w (= exp(log w))
}

// ---- inclusive prefix sum of w (single block of 1024, 8 elems each) ----
__global__ void __launch_bounds__(1024) k_scan(float* __restrict__ ws) {
    __shared__ float ls[1024];
    int t = threadIdx.x;
    float seg[8];
    float sum = 0.0f;
    #pragma unroll
    for (int e = 0; e < 8; ++e) { sum += ws[OFF_W + t * 8 + e]; seg[e] = sum; }
    ls[t] = sum;
    __syncthreads();
    for (int off = 1; off < 1024; off <<= 1) {
        float add = (t >= off) ? ls[t - off] : 0.0f;
        __syncthreads();
        ls[t] += add;
        __syncthreads();
    }
    float base = (t > 0) ? ls[t - 1] : 0.0f;
    #pragma unroll
    for (int e = 0; e < 8; ++e) ws[OFF_PREF + t * 8 + e] = base + seg[e];
    if (t == 1023) ws[OFF_WTOT] = ls[1023];
}

// ---- categorical resample (inverse CDF) + accumulate mean ----
__global__ void k_resample(float* __restrict__ ws) {
    int k = blockIdx.x * blockDim.x + threadIdx.x;
    if (k >= PN) return;
    float W = ws[OFF_WTOT];
    float u = u01(pcg_hash(0x9E3779B9u ^ (unsigned)k)) * W;
    const float* pref = ws + OFF_PREF;
    int lo = 0, hi = PN - 1;
    while (lo < hi) {
        int mid = (lo + hi) >> 1;
        if (pref[mid] < u) lo = mid + 1; else hi = mid;
    }
    const float* st = ws + OFF_STATE;
    atomicAdd(&ws[OFF_ACC + 0], st[lo * 3 + 0]);
    atomicAdd(&ws[OFF_ACC + 1], st[lo * 3 + 1]);
    atomicAdd(&ws[OFF_ACC + 2], st[lo * 3 + 2]);
}

__global__ void k_final(const float* __restrict__ ws, float* __restrict__ out) {
    int c = threadIdx.x;
    if (c < 3) out[c] = ws[OFF_ACC + c] * (1.0f / (float)PN);
}

extern "C" void kernel_launch(void* const* d_in, const int* in_sizes, int n_in,
                              void* d_out, int out_size, void* d_ws, size_t ws_size,
                              hipStream_t stream) {
    const float* inputs = (const float*)d_in[0];   // (1, C, P)
    const float* sv     = (const float*)d_in[1];   // (P, 3)
    const float* T      = (const float*)d_in[2];   // (3, 3)
    const float* Q      = (const float*)d_in[3];   // (3, 3)
    const float* fwd    = (const float*)d_in[4];   // (C, 3)
    float*       out    = (float*)d_out;           // (3,)
    float*       ws     = (float*)d_ws;

    k_zero<<<dim3(128), dim3(256), 0, stream>>>(ws);
    k_state<<<dim3(PN / 256), dim3(256), 0, stream>>>(sv, T, Q, ws);
    k_gram<<<dim3(32), dim3(256), 0, stream>>>(fwd, ws);
    // rows: 512 tiles of 16, 4 waves/block -> 128 blocks; K: 8 chunks of 1024
    k_main<<<dim3(128, 8), dim3(128), 0, stream>>>(inputs, fwd, ws);
    k_weights<<<dim3(PN / 256), dim3(256), 0, stream>>>(ws);
    k_scan<<<dim3(1), dim3(1024), 0, stream>>>(ws);
    k_resample<<<dim3(PN / 256), dim3(256), 0, stream>>>(ws);
    k_final<<<dim3(1), dim3(32), 0, stream>>>(ws, out);
}